// LSTMCell_47296179864168
// MI455X (gfx1250) — compile-verified
//
#include <hip/hip_runtime.h>

typedef __attribute__((ext_vector_type(2))) float v2f;
typedef __attribute__((ext_vector_type(8))) float v8f;

#define TILE_ROWS 128   // rows per workgroup
#define LDA 260         // Abuf row stride in floats (256 + 4 pad)
#define LDP 288         // Wp pair-row stride in floats (256 + 32 pad -> 32-bank shift)
#define N_COLS 128
#define K_TOT 256       // concatenated K ([input|hidden] / [cell|hidden])

__device__ __forceinline__ float sigmoid_fast(float x) {
    return 1.0f / (1.0f + __expf(-x));
}

// Stage a 256x128 row-major weight matrix into LDS in K-pair-interleaved form:
//   Wp[p*LDP + 2*c + 0] = W[2p][c],  Wp[p*LDP + 2*c + 1] = W[2p+1][c]
// so a B-fragment (rows k,k+1 at one column) is a single aligned 8-byte load.
__device__ __forceinline__ void stage_weight(const float* __restrict__ w,
                                             float* __restrict__ Wp, int t) {
    for (int q = t; q < 128 * 32; q += 256) {
        const int p = q >> 5;              // pair row 0..127  (K rows 2p, 2p+1)
        const int c = (q & 31) * 4;        // col 0,4,...,124
        const float4 r0 = *(const float4*)&w[(size_t)(2 * p)     * N_COLS + c];
        const float4 r1 = *(const float4*)&w[(size_t)(2 * p + 1) * N_COLS + c];
        const float4 lo = {r0.x, r1.x, r0.y, r1.y};
        const float4 hi = {r0.z, r1.z, r0.w, r1.w};
        *(float4*)&Wp[p * LDP + 2 * c]     = lo;
        *(float4*)&Wp[p * LDP + 2 * c + 4] = hi;
    }
}

// One GEMM stage: acc(16x128 per wave) += A(16x256) @ W(256x128), K-tiled by 4.
__device__ __forceinline__ void gemm_stage(const float* __restrict__ Abuf,
                                           const float* __restrict__ Wp,
                                           int aRow, int lhi, int llo,
                                           v8f acc[8]) {
    const float* __restrict__ wcol = Wp + 2 * llo;   // this lane's column base
    for (int kt = 0; kt < K_TOT / 4; ++kt) {
        // A frag: lanes 0-15 hold K={4kt,4kt+1}, lanes 16-31 K={4kt+2,4kt+3}
        const v2f a = *(const v2f*)&Abuf[aRow * LDA + kt * 4 + 2 * lhi];
        const float* __restrict__ wrow = wcol + (kt * 2 + lhi) * LDP;
        #pragma unroll
        for (int n = 0; n < 8; ++n) {
            const v2f b = *(const v2f*)&wrow[n * 32];  // (W[k][c], W[k+1][c])
            acc[n] = __builtin_amdgcn_wmma_f32_16x16x4_f32(
                false, a, false, b, (short)0, acc[n], false, false);
        }
    }
}

__global__ __launch_bounds__(256, 1)
void lstm_cell_wmma_f32(const float* __restrict__ input,
                        const float* __restrict__ hidden,
                        const float* __restrict__ w_ih,
                        const float* __restrict__ w_hh,
                        const float* __restrict__ b_ih,
                        const float* __restrict__ b_hh,
                        float* __restrict__ out)
{
    extern __shared__ float smem[];
    float* Abuf = smem;                      // 128 x LDA ([input|hidden] -> [cell|hidden])
    float* Wbuf = smem + TILE_ROWS * LDA;    // 128 pair-rows x LDP (w_ih -> w_hh)

    const int t    = threadIdx.x;
    const int lane = t & 31;                 // wave32
    const int wv   = t >> 5;                 // wave id 0..7
    const int llo  = lane & 15;
    const int lhi  = lane >> 4;              // 0 or 1

    const int blockRow0 = blockIdx.x * TILE_ROWS;

    // ---- Cooperative staging: input|hidden rows -> Abuf, weight_ih -> Wbuf
    for (int q = t; q < (TILE_ROWS * N_COLS) / 4; q += 256) {
        const int row = q >> 5;              // 32 float4 per 128-col row
        const int col = (q & 31) * 4;
        const float4 vi = *(const float4*)&input [(size_t)(blockRow0 + row) * N_COLS + col];
        const float4 vh = *(const float4*)&hidden[(size_t)(blockRow0 + row) * N_COLS + col];
        *(float4*)&Abuf[row * LDA + col]       = vi;
        *(float4*)&Abuf[row * LDA + 128 + col] = vh;
    }
    stage_weight(w_ih, Wbuf, t);
    __syncthreads();

    const int waveRow0 = wv * 16;            // this wave's 16-row stripe
    const int aRow     = waveRow0 + llo;     // A-frag row for this lane
    const int cRow0    = waveRow0 + 8 * lhi; // C/D layout: first of the 8 rows per lane

    // ---- Stage 1: acc = [input|hidden] @ weight_ih
    v8f acc[8] = {};
    gemm_stage(Abuf, Wbuf, aRow, lhi, llo, acc);
    __syncthreads();   // all waves done reading Wbuf(w_ih) and Abuf cell region

    // ---- cell = sigmoid(acc + bias_ih); write into Abuf cols [0,128)
    #pragma unroll
    for (int n = 0; n < 8; ++n) {
        const float bi = b_ih[n * 16 + llo];
        #pragma unroll
        for (int i = 0; i < 8; ++i) {
            Abuf[(cRow0 + i) * LDA + n * 16 + llo] = sigmoid_fast(acc[n][i] + bi);
        }
    }
    // ---- swap weights: weight_hh -> Wbuf
    stage_weight(w_hh, Wbuf, t);
    __syncthreads();

    // ---- Stage 2: acc2 = [cell|hidden] @ weight_hh
    v8f acc2[8] = {};
    gemm_stage(Abuf, Wbuf, aRow, lhi, llo, acc2);

    // ---- hidden_out = sigmoid(acc2 + bias_hh) -> global
    const size_t gRow0 = (size_t)blockRow0 + (size_t)cRow0;
    #pragma unroll
    for (int n = 0; n < 8; ++n) {
        const float bh = b_hh[n * 16 + llo];
        #pragma unroll
        for (int i = 0; i < 8; ++i) {
            out[(gRow0 + i) * N_COLS + n * 16 + llo] = sigmoid_fast(acc2[n][i] + bh);
        }
    }
}

extern "C" void kernel_launch(void* const* d_in, const int* in_sizes, int n_in,
                              void* d_out, int out_size, void* d_ws, size_t ws_size,
                              hipStream_t stream) {
    const float* input  = (const float*)d_in[0];
    const float* hidden = (const float*)d_in[1];
    const float* w_ih   = (const float*)d_in[2];
    const float* w_hh   = (const float*)d_in[3];
    const float* b_ih   = (const float*)d_in[4];
    const float* b_hh   = (const float*)d_in[5];
    float* out = (float*)d_out;

    const int rows = in_sizes[0] / N_COLS;          // B = 524288
    const int grid = rows / TILE_ROWS;              // 4096 blocks
    const size_t shmem = (size_t)(TILE_ROWS * LDA + 128 * LDP) * sizeof(float); // 280,576 B

    lstm_cell_wmma_f32<<<grid, 256, shmem, stream>>>(input, hidden, w_ih, w_hh,
                                                     b_ih, b_hh, out);
}